// SimFeatureSearch_86663850099177
// MI455X (gfx1250) — compile-verified
//
#include <hip/hip_runtime.h>
#include <hip/hip_bf16.h>

typedef __attribute__((ext_vector_type(2))) float v2f;
typedef __attribute__((ext_vector_type(8))) float v8f;

// Problem constants (from reference): feature_map (N=2, C=64, D=16, H=32, W=32)
// D smallest -> slice along depth. X(s,n) is (C, L=H*W=1024), channel stride D*L.
static constexpr int kN = 2;
static constexpr int kC = 64;
static constexpr int kD = 16;
static constexpr int kL = 1024;          // H*W
static constexpr int kChStride = kD * kL; // stride between channels inside X view
static constexpr int kTopK = 5;

__launch_bounds__(128)
__global__ void simfeat_topk_kernel(const float* __restrict__ fm,
                                    float* __restrict__ out) {
    // 16-row sim block for this workgroup (padded row stride to spread banks)
    __shared__ float sim[16][kL + 1];
    __shared__ float cand_v[16][8][kTopK];
    __shared__ int   cand_i[16][8][kTopK];
    __shared__ float wv[16][kTopK];
    __shared__ int   wi[16][kTopK];

    const int blk = blockIdx.x;          // 0 .. 16*2*64-1
    const int rb  = blk & 63;            // row block within L
    const int n   = (blk >> 6) & 1;
    const int s   = blk >> 7;            // depth slice
    const int row_base = rb * 16;

    // X(c, l) = fm[((n*C + c)*D + s)*L + l]
    const float* __restrict__ X =
        fm + (size_t)n * kC * kD * kL + (size_t)s * kL;

    const int tid  = threadIdx.x;
    const int lane = tid & 31;
    const int wave = tid >> 5;           // 4 waves
    const int half = lane >> 4;          // 0: lanes 0-15, 1: lanes 16-31
    const int lp   = lane & 15;

    // ---------------- Phase 1: Gram tiles via V_WMMA_F32_16X16X4_F32 -------
    // A (16x64): per K-chunk of 4 channels, lane layout (ISA 7.12.2):
    //   lanes 0-15 hold K = c0, c0+1 ; lanes 16-31 hold K = c0+2, c0+3
    v2f A[16];
#pragma unroll
    for (int cc = 0; cc < 16; ++cc) {
        const int c0 = cc * 4 + half * 2;
        A[cc].x = X[(size_t)(c0    ) * kChStride + row_base + lp];
        A[cc].y = X[(size_t)(c0 + 1) * kChStride + row_base + lp];
    }

    // Each wave computes 16 of the 64 column tiles.
    for (int t = 0; t < 16; ++t) {
        const int col_base = (wave * 16 + t) * 16;
        v8f acc = {};
#pragma unroll
        for (int cc = 0; cc < 16; ++cc) {
            const int c0 = cc * 4 + half * 2;
            v2f B;
            B.x = X[(size_t)(c0    ) * kChStride + col_base + lp];
            B.y = X[(size_t)(c0 + 1) * kChStride + col_base + lp];
            acc = __builtin_amdgcn_wmma_f32_16x16x4_f32(
                /*neg_a=*/false, A[cc], /*neg_b=*/false, B,
                /*c_mod=*/(short)0, acc, /*reuse_a=*/false, /*reuse_b=*/false);
        }
        // C/D layout: VGPR r -> row r (lanes 0-15) / row r+8 (lanes 16-31),
        // column = col_base + (lane & 15).
#pragma unroll
        for (int r = 0; r < 8; ++r)
            sim[r + half * 8][col_base + lp] = acc[r];
    }
    __syncthreads();

    // ---------------- Phase 2: per-row top-5 -------------------------------
    {
        const int row = tid >> 3;        // 16 rows
        const int seg = tid & 7;         // 8 scanners per row, stride-8 columns
        float bv[kTopK];
        int   bi[kTopK];
#pragma unroll
        for (int k = 0; k < kTopK; ++k) { bv[k] = -INFINITY; bi[k] = 0x7fffffff; }
        for (int j = 0; j < kL / 8; ++j) {
            const int col = seg + j * 8;     // increasing -> earliest index wins ties
            const float v = sim[row][col];
            if (v > bv[kTopK - 1]) {
                int p = kTopK - 1;
                while (p > 0 && v > bv[p - 1]) {
                    bv[p] = bv[p - 1]; bi[p] = bi[p - 1]; --p;
                }
                bv[p] = v; bi[p] = col;
            }
        }
#pragma unroll
        for (int k = 0; k < kTopK; ++k) {
            cand_v[row][seg][k] = bv[k];
            cand_i[row][seg][k] = bi[k];
        }
    }
    __syncthreads();

    // Merge the 8 sorted candidate lists per row; tie-break lower index first
    if (tid < 16) {
        const int row = tid;
        int p[8] = {0, 0, 0, 0, 0, 0, 0, 0};
        int   ti[kTopK];
        for (int k = 0; k < kTopK; ++k) {
            float best = -INFINITY; int besti = 0x7fffffff; int bestseg = 0;
#pragma unroll
            for (int sgm = 0; sgm < 8; ++sgm) {
                if (p[sgm] < kTopK) {
                    const float v = cand_v[row][sgm][p[sgm]];
                    const int   i = cand_i[row][sgm][p[sgm]];
                    if (v > best || (v == best && i < besti)) {
                        best = v; besti = i; bestseg = sgm;
                    }
                }
            }
            p[bestseg]++;
            ti[k] = besti;
        }
        // inverse-distance weights, normalized over k
        const float pos = (float)(row_base + row);
        float w[kTopK], sum = 0.f;
#pragma unroll
        for (int k = 0; k < kTopK; ++k) {
            const float dist = fabsf((float)ti[k] - pos) + 1e-5f;
            w[k] = 1.0f / dist;
            sum += w[k];
        }
#pragma unroll
        for (int k = 0; k < kTopK; ++k) {
            wv[row][k] = w[k] / sum;
            wi[row][k] = ti[k];
        }
    }
    __syncthreads();

    // ---------------- Phase 3: weighted gather over channels ---------------
    // Output is a raw reshape of wavg (S,N,C,L):
    //   out[((s*2 + n)*64 + c)*1024 + l]
    const size_t obase = (((size_t)s * kN + n) * kC) * kL;
#pragma unroll
    for (int it = 0; it < 8; ++it) {
        const int o = tid + it * 128;    // 0 .. 1023
        const int c = o >> 4;
        const int r = o & 15;
        float accv = 0.f;
#pragma unroll
        for (int k = 0; k < kTopK; ++k)
            accv += wv[r][k] * X[(size_t)c * kChStride + wi[r][k]];
        out[obase + (size_t)c * kL + row_base + r] = accv;
    }
}

extern "C" void kernel_launch(void* const* d_in, const int* in_sizes, int n_in,
                              void* d_out, int out_size, void* d_ws, size_t ws_size,
                              hipStream_t stream) {
    (void)in_sizes; (void)n_in; (void)d_ws; (void)ws_size; (void)out_size;
    const float* fm = (const float*)d_in[0];
    float* out = (float*)d_out;
    // one block per (s, n, 16-row block): 16 * 2 * 64 = 2048 blocks, 128 threads
    simfeat_topk_kernel<<<dim3(kD * kN * (kL / 16)), dim3(128), 0, stream>>>(fm, out);
}